// BayesLinear_50929722196753
// MI455X (gfx1250) — compile-verified
//
#include <hip/hip_runtime.h>
#include <math.h>

typedef float v2f __attribute__((ext_vector_type(2)));
typedef float v4f __attribute__((ext_vector_type(4)));
typedef float v8f __attribute__((ext_vector_type(8)));

#define B_N   1024
#define IN_N  512
#define OUT_N 512

// ---------------------------------------------------------------------------
// Kernel 1: sigma[o,i] = exp(weight_psi[o,i])   (1 MB, will live in L2)
// ---------------------------------------------------------------------------
__global__ void __launch_bounds__(256) prep_sigma(const float* __restrict__ psi,
                                                  float* __restrict__ sigma) {
    int i = blockIdx.x * 256 + threadIdx.x;
    sigma[i] = expf(psi[i]);
}

// ---------------------------------------------------------------------------
// Kernel 2: out[b,o] = sum_i x[b,i]*mu[o,i]  + eps_b[b,o]*exp(bias_psi[o]) + bias_mu[o]
// WMMA f32 16x16x4, D = A(16b x 4i) * B(4i x 16o) + C, one 16x16 tile per wave.
// M = batch rows, N = out cols  -> stores coalesced along o.
// ---------------------------------------------------------------------------
__global__ void __launch_bounds__(256) gemm_bias(const float* __restrict__ x,
                                                 const float* __restrict__ mu,
                                                 const float* __restrict__ eps_b,
                                                 const float* __restrict__ bias_mu,
                                                 const float* __restrict__ bias_psi,
                                                 float* __restrict__ out) {
    const int wave = threadIdx.x >> 5;
    const int lane = threadIdx.x & 31;
    const int tile = blockIdx.x * 8 + wave;      // 2048 tiles = 64 b-tiles * 32 o-tiles
    const int B0 = (tile >> 5) * 16;             // batch tile origin
    const int O0 = (tile & 31) * 16;             // out tile origin
    const int half = lane >> 4;                  // lanes 16..31 carry K+2,K+3
    const int mn   = lane & 15;                  // A row (batch) and B col (out) index

    const float* __restrict__ xrow  = x  + (size_t)(B0 + mn) * IN_N + 2 * half;
    const float* __restrict__ murow = mu + (size_t)(O0 + mn) * IN_N + 2 * half;

    v8f c = {0.f, 0.f, 0.f, 0.f, 0.f, 0.f, 0.f, 0.f};
    #pragma unroll 8
    for (int k = 0; k < IN_N; k += 4) {
        v2f a = { xrow[k],  xrow[k + 1]  };      // A: x tile, K window {k..k+3}
        v2f b = { murow[k], murow[k + 1] };      // B: mu^T tile
        c = __builtin_amdgcn_wmma_f32_16x16x4_f32(
                /*neg_a=*/false, a, /*neg_b=*/false, b,
                /*c_mod=*/(short)0, c, /*reuse_a=*/false, /*reuse_b=*/false);
    }

    // Epilogue: o is fixed per lane, so the bias exp is computed once per lane.
    const int   o  = O0 + mn;
    const float sb = expf(bias_psi[o]);
    const float bm = bias_mu[o];
    #pragma unroll
    for (int r = 0; r < 8; ++r) {
        const int b = B0 + r + half * 8;         // C/D layout: VGPR r -> row r (+8 upper half)
        const size_t idx = (size_t)b * OUT_N + o;
        out[idx] = c[r] + eps_b[idx] * sb + bm;
    }
}

// ---------------------------------------------------------------------------
// Kernel 3: out[b,o] += sum_i eps_w[b,o,i] * sigma[o,i] * x[b,i]
// One wave per (b,o). eps row = 2 KB streamed once (non-temporal b128 loads,
// fully coalesced: 32 lanes x float4 = 512 B per load, 4 loads per row).
// sigma/x rows are L2-hot (reused 1024x / 512x).
// ---------------------------------------------------------------------------
__global__ void __launch_bounds__(256) eps_stream(const float* __restrict__ eps_w,
                                                  const float* __restrict__ sigma,
                                                  const float* __restrict__ x,
                                                  float* __restrict__ out) {
    const int wid  = (blockIdx.x * 256 + threadIdx.x) >> 5;  // 0 .. 524287
    const int lane = threadIdx.x & 31;
    const int b = wid >> 9;
    const int o = wid & 511;

    const v4f* __restrict__ e4 = (const v4f*)(eps_w + ((size_t)b * OUT_N + o) * IN_N);
    const v4f* __restrict__ s4 = (const v4f*)(sigma + (size_t)o * IN_N);
    const v4f* __restrict__ x4 = (const v4f*)(x + (size_t)b * IN_N);

    float acc = 0.f;
    #pragma unroll
    for (int j = 0; j < 4; ++j) {
        const int idx = j * 32 + lane;                       // float4 index within row
        v4f e  = __builtin_nontemporal_load(&e4[idx]);       // streaming, no reuse
        v4f s  = s4[idx];
        v4f xv = x4[idx];
        acc = fmaf(e[0] * s[0], xv[0], acc);
        acc = fmaf(e[1] * s[1], xv[1], acc);
        acc = fmaf(e[2] * s[2], xv[2], acc);
        acc = fmaf(e[3] * s[3], xv[3], acc);
    }

    // wave32 butterfly reduction
    #pragma unroll
    for (int m = 16; m >= 1; m >>= 1)
        acc += __shfl_xor(acc, m, 32);

    if (lane == 0)
        out[(size_t)b * OUT_N + o] += acc;                   // out was fully written by gemm_bias
}

// ---------------------------------------------------------------------------
extern "C" void kernel_launch(void* const* d_in, const int* in_sizes, int n_in,
                              void* d_out, int out_size, void* d_ws, size_t ws_size,
                              hipStream_t stream) {
    const float* x     = (const float*)d_in[0];   // [B, IN]
    const float* eps_w = (const float*)d_in[1];   // [B, OUT, IN]
    const float* eps_b = (const float*)d_in[2];   // [B, OUT]
    const float* wmu   = (const float*)d_in[3];   // [OUT, IN]
    const float* wpsi  = (const float*)d_in[4];   // [OUT, IN]
    const float* bmu   = (const float*)d_in[5];   // [OUT]
    const float* bpsi  = (const float*)d_in[6];   // [OUT]
    float* out   = (float*)d_out;                 // [B, OUT]
    float* sigma = (float*)d_ws;                  // OUT*IN floats = 1 MB scratch

    prep_sigma<<<(OUT_N * IN_N) / 256, 256, 0, stream>>>(wpsi, sigma);
    gemm_bias <<<(64 * 32) / 8,       256, 0, stream>>>(x, wmu, eps_b, bmu, bpsi, out);
    eps_stream<<<(B_N * OUT_N) / 8,   256, 0, stream>>>(eps_w, sigma, x, out);
}